// COOLCHIC_INTERP_ENC_27075473834477
// MI455X (gfx1250) — compile-verified
//
#include <hip/hip_runtime.h>

#define DEVINL __device__ __forceinline__

typedef float vf4 __attribute__((ext_vector_type(4)));
typedef float vf2 __attribute__((ext_vector_type(2)));

constexpr int H_OUT = 721;
constexpr int W_OUT = 1440;
constexpr int NLEV  = 8;

// Pyramid level dims: ceil(721/2^i) x ceil(1440/2^i)
// i:      0     1    2    3   4   5   6   7
// H:    721   361  181   91  46  23  12   6
// W:   1440   720  360  180  90  45  23  12

// LDS staging for levels 4..7 (5523 floats = 21.6 KB).
// Layout order 4,6,7,5 keeps every region 16B-aligned for b128 async loads:
//   O4 = 0      (4140 floats, 16560 B)
//   O6 = 4140   (276  floats)
//   O7 = 4416   (72   floats)
//   O5 = 4488   (1035 floats)
constexpr int LDS_N4 = 46 * 90;   // 4140
constexpr int LDS_N5 = 23 * 45;   // 1035
constexpr int LDS_N6 = 12 * 23;   // 276
constexpr int LDS_N7 = 6  * 12;   // 72
constexpr int LDS_O4 = 0;
constexpr int LDS_O6 = LDS_O4 + LDS_N4;   // 4140
constexpr int LDS_O7 = LDS_O6 + LDS_N6;   // 4416
constexpr int LDS_O5 = LDS_O7 + LDS_N7;   // 4488
constexpr int LDS_TOT = LDS_O5 + LDS_N5;  // 5523 floats

// ---------------------------------------------------------------------------
// CDNA5 async global->LDS staging (ASYNCcnt-tracked).
// VDST = LDS byte address (generic LDS pointers carry the workgroup-relative
// offset in their low 32 bits), VADDR = 64-bit global address, saddr = off.
// Bulk in b128, tail in b32.
// ---------------------------------------------------------------------------
DEVINL void lds_stage_async(float* dst, const float* __restrict__ src, int count) {
    const int tid = (int)threadIdx.x;
    const int nth = (int)blockDim.x;
    const int n4  = count >> 2;
    for (int i = tid; i < n4; i += nth) {
        unsigned           ldsAddr = (unsigned)(size_t)(dst + i * 4);
        unsigned long long gAddr   = (unsigned long long)(size_t)(src + i * 4);
        asm volatile("global_load_async_to_lds_b128 %0, %1, off"
                     :: "v"(ldsAddr), "v"(gAddr)
                     : "memory");
    }
    for (int i = (n4 << 2) + tid; i < count; i += nth) {
        unsigned           ldsAddr = (unsigned)(size_t)(dst + i);
        unsigned long long gAddr   = (unsigned long long)(size_t)(src + i);
        asm volatile("global_load_async_to_lds_b32 %0, %1, off"
                     :: "v"(ldsAddr), "v"(gAddr)
                     : "memory");
    }
}

DEVINL void wait_async_then_barrier() {
#if __has_builtin(__builtin_amdgcn_s_wait_asynccnt)
    __builtin_amdgcn_s_wait_asynccnt(0);
#else
    asm volatile("s_wait_asynccnt 0" ::: "memory");
#endif
    __syncthreads();
}

// ---------------------------------------------------------------------------
// Upsample axis parameterization, matching the jax reference exactly:
//   s = clip((o+0.5)*(n_in/n_out) - 0.5, 0, n_in-1); i0=floor(s);
//   i1=min(i0+1, n_in-1); w = s - i0
// ---------------------------------------------------------------------------
DEVINL void axis_param(int o, float scale, int n_in, int& i0, int& i1, float& w) {
    float s = fminf(fmaxf(((float)o + 0.5f) * scale - 0.5f, 0.0f), (float)(n_in - 1));
    i0 = (int)s;                    // s >= 0, truncation == floor
    w  = s - (float)i0;
    i1 = (i0 + 1 < n_in) ? (i0 + 1) : (n_in - 1);
}

DEVINL float lerp2(const float* __restrict__ g, int Wl,
                   int r0, int r1, float wr, int c0, int c1, float wc) {
    float a = g[r0 * Wl + c0] * (1.0f - wc) + g[r0 * Wl + c1] * wc;
    float b = g[r1 * Wl + c0] * (1.0f - wc) + g[r1 * Wl + c1] * wc;
    return a * (1.0f - wr) + b * wr;
}

DEVINL float up_bilerp(const float* __restrict__ g, int Hl, int Wl, int r, int c) {
    const float sh = (float)Hl * (1.0f / 721.0f);
    const float sw = (float)Wl * (1.0f / 1440.0f);
    int r0, r1, c0, c1; float wr, wc;
    axis_param(r, sh, Hl, r0, r1, wr);
    axis_param(c, sw, Wl, c0, c1, wc);
    return lerp2(g, Wl, r0, r1, wr, c0, c1, wc);
}

// ---------------------------------------------------------------------------
// Kernel 1: materialize all 8 upsampled levels, level-innermost layout:
//   ws[(r*1440 + c)*8 + l]  -> two b128 stores per pixel (stays L2-resident)
// Levels 4..7 gathered from LDS (async-staged), 1..3 from L2-resident global,
// level 0 is the identity upsample (scale == 1).
// ---------------------------------------------------------------------------
__global__ void __launch_bounds__(256)
coolchic_up_pack_kernel(const float* __restrict__ g0, const float* __restrict__ g1,
                        const float* __restrict__ g2, const float* __restrict__ g3,
                        const float* __restrict__ g4, const float* __restrict__ g5,
                        const float* __restrict__ g6, const float* __restrict__ g7,
                        float* __restrict__ ws) {
    __shared__ float smem[LDS_TOT];
    lds_stage_async(smem + LDS_O4, g4, LDS_N4);
    lds_stage_async(smem + LDS_O6, g6, LDS_N6);
    lds_stage_async(smem + LDS_O7, g7, LDS_N7);
    lds_stage_async(smem + LDS_O5, g5, LDS_N5);
    wait_async_then_barrier();

    const int idx = (int)(blockIdx.x * blockDim.x + threadIdx.x);
    if (idx >= H_OUT * W_OUT) return;
    const int r = idx / W_OUT;
    const int c = idx - r * W_OUT;

    float v0 = g0[idx];  // identity upsample at level 0
    float v1 = up_bilerp(g1, 361, 720, r, c);
    float v2 = up_bilerp(g2, 181, 360, r, c);
    float v3 = up_bilerp(g3,  91, 180, r, c);
    float v4 = up_bilerp(smem + LDS_O4, 46, 90, r, c);
    float v5 = up_bilerp(smem + LDS_O5, 23, 45, r, c);
    float v6 = up_bilerp(smem + LDS_O6, 12, 23, r, c);
    float v7 = up_bilerp(smem + LDS_O7,  6, 12, r, c);

    vf4* w4 = (vf4*)(ws + (size_t)idx * 8);
    w4[0] = (vf4){v0, v1, v2, v3};   // regular-temporal: next kernel reads from L2
    w4[1] = (vf4){v4, v5, v6, v7};
}

// ---------------------------------------------------------------------------
// Kernel 2: per-point sample. Corner indices/fracs are identical across all 8
// levels, so each of the 4 corners is one 32B (2 x b128) fetch from the packed
// L2-resident intermediate. x loads and out stores are non-temporal to keep
// the 33MB intermediate + pyramid resident in L2.
// Matches reference: frac is relative to the CLAMPED floor.
// ---------------------------------------------------------------------------
__global__ void __launch_bounds__(256)
coolchic_sample_pack_kernel(const float* __restrict__ x, const float* __restrict__ ws,
                            float* __restrict__ out, int n) {
    const int i = (int)(blockIdx.x * blockDim.x + threadIdx.x);
    if (i >= n) return;
    vf2 p = __builtin_nontemporal_load((const vf2*)x + i);
    float lat_idx = (90.0f - p.x) * 4.0f;  // /0.25
    float lon_idx = p.y * 4.0f;

    float flr = floorf(lat_idx);
    float flc = floorf(lon_idx);
    float lat_f = fminf(fmaxf(flr,        0.0f), 720.0f);
    float lat_c = fminf(fmaxf(flr + 1.0f, 0.0f), 720.0f);
    float lon_f = fminf(fmaxf(flc,        0.0f), 1439.0f);
    float lon_c = fminf(fmaxf(flc + 1.0f, 0.0f), 1439.0f);
    int lf = (int)lat_f, lc = (int)lat_c, jf = (int)lon_f, jc = (int)lon_c;
    float fr = lat_idx - lat_f;
    float fc = lon_idx - lon_f;

    const vf4* w4 = (const vf4*)ws;
    int iff = (lf * W_OUT + jf) * 2;
    int ifc = (lf * W_OUT + jc) * 2;
    int icf = (lc * W_OUT + jf) * 2;
    int icc = (lc * W_OUT + jc) * 2;

    float vff[8], vfc[8], vcf[8], vcc[8];
    *(vf4*)&vff[0] = w4[iff]; *(vf4*)&vff[4] = w4[iff + 1];
    *(vf4*)&vfc[0] = w4[ifc]; *(vf4*)&vfc[4] = w4[ifc + 1];
    *(vf4*)&vcf[0] = w4[icf]; *(vf4*)&vcf[4] = w4[icf + 1];
    *(vf4*)&vcc[0] = w4[icc]; *(vf4*)&vcc[4] = w4[icc + 1];

    float o[8];
#pragma unroll
    for (int l = 0; l < NLEV; ++l) {
        float vf = vff[l] + fc * (vfc[l] - vff[l]);
        float vc = vcf[l] + fc * (vcc[l] - vcf[l]);
        o[l] = vf + fr * (vc - vf);
    }
    vf4* o4 = (vf4*)(out + (size_t)i * 8);
    __builtin_nontemporal_store(*(vf4*)&o[0], o4);
    __builtin_nontemporal_store(*(vf4*)&o[4], o4 + 1);
}

// ---------------------------------------------------------------------------
// Fallback: fully fused (no workspace). Composes the two bilerps exactly:
// per level, 4 axis parameterizations shared across the 16 taps.
// ---------------------------------------------------------------------------
DEVINL float level_sample(const float* __restrict__ g, int Hl, int Wl,
                          int lf, int lc, int jf, int jc, float fr, float fc) {
    const float sh = (float)Hl * (1.0f / 721.0f);
    const float sw = (float)Wl * (1.0f / 1440.0f);
    int rf0, rf1, rc0, rc1, cf0, cf1, cc0, cc1;
    float wrf, wrc, wcf, wcc;
    axis_param(lf, sh, Hl, rf0, rf1, wrf);
    axis_param(lc, sh, Hl, rc0, rc1, wrc);
    axis_param(jf, sw, Wl, cf0, cf1, wcf);
    axis_param(jc, sw, Wl, cc0, cc1, wcc);
    float vff = lerp2(g, Wl, rf0, rf1, wrf, cf0, cf1, wcf);
    float vfc = lerp2(g, Wl, rf0, rf1, wrf, cc0, cc1, wcc);
    float vcf = lerp2(g, Wl, rc0, rc1, wrc, cf0, cf1, wcf);
    float vcc = lerp2(g, Wl, rc0, rc1, wrc, cc0, cc1, wcc);
    float vf = vff + fc * (vfc - vff);
    float vc = vcf + fc * (vcc - vcf);
    return vf + fr * (vc - vf);
}

__global__ void __launch_bounds__(256)
coolchic_fused_kernel(const float* __restrict__ x,
                      const float* __restrict__ g0, const float* __restrict__ g1,
                      const float* __restrict__ g2, const float* __restrict__ g3,
                      const float* __restrict__ g4, const float* __restrict__ g5,
                      const float* __restrict__ g6, const float* __restrict__ g7,
                      float* __restrict__ out, int n) {
    __shared__ float smem[LDS_TOT];
    lds_stage_async(smem + LDS_O4, g4, LDS_N4);
    lds_stage_async(smem + LDS_O6, g6, LDS_N6);
    lds_stage_async(smem + LDS_O7, g7, LDS_N7);
    lds_stage_async(smem + LDS_O5, g5, LDS_N5);
    wait_async_then_barrier();

    const int i = (int)(blockIdx.x * blockDim.x + threadIdx.x);
    if (i >= n) return;
    vf2 p = __builtin_nontemporal_load((const vf2*)x + i);
    float lat_idx = (90.0f - p.x) * 4.0f;
    float lon_idx = p.y * 4.0f;
    float flr = floorf(lat_idx);
    float flc = floorf(lon_idx);
    float lat_f = fminf(fmaxf(flr,        0.0f), 720.0f);
    float lat_c = fminf(fmaxf(flr + 1.0f, 0.0f), 720.0f);
    float lon_f = fminf(fmaxf(flc,        0.0f), 1439.0f);
    float lon_c = fminf(fmaxf(flc + 1.0f, 0.0f), 1439.0f);
    int lf = (int)lat_f, lc = (int)lat_c, jf = (int)lon_f, jc = (int)lon_c;
    float fr = lat_idx - lat_f;
    float fc = lon_idx - lon_f;

    float o[8];
    {   // level 0: upsample is identity
        float vff = g0[lf * W_OUT + jf];
        float vfc = g0[lf * W_OUT + jc];
        float vcf = g0[lc * W_OUT + jf];
        float vcc = g0[lc * W_OUT + jc];
        float vf = vff + fc * (vfc - vff);
        float vc = vcf + fc * (vcc - vcf);
        o[0] = vf + fr * (vc - vf);
    }
    o[1] = level_sample(g1, 361, 720, lf, lc, jf, jc, fr, fc);
    o[2] = level_sample(g2, 181, 360, lf, lc, jf, jc, fr, fc);
    o[3] = level_sample(g3,  91, 180, lf, lc, jf, jc, fr, fc);
    o[4] = level_sample(smem + LDS_O4, 46, 90, lf, lc, jf, jc, fr, fc);
    o[5] = level_sample(smem + LDS_O5, 23, 45, lf, lc, jf, jc, fr, fc);
    o[6] = level_sample(smem + LDS_O6, 12, 23, lf, lc, jf, jc, fr, fc);
    o[7] = level_sample(smem + LDS_O7,  6, 12, lf, lc, jf, jc, fr, fc);

    vf4* o4 = (vf4*)(out + (size_t)i * 8);
    __builtin_nontemporal_store(*(vf4*)&o[0], o4);
    __builtin_nontemporal_store(*(vf4*)&o[4], o4 + 1);
}

extern "C" void kernel_launch(void* const* d_in, const int* in_sizes, int n_in,
                              void* d_out, int out_size, void* d_ws, size_t ws_size,
                              hipStream_t stream) {
    const float* x = (const float*)d_in[0];
    const float* g[NLEV];
    for (int l = 0; l < NLEV; ++l) g[l] = (const float*)d_in[1 + l];
    float* out = (float*)d_out;
    const int n = in_sizes[0] / 2;

    const size_t need = (size_t)H_OUT * W_OUT * NLEV * sizeof(float);  // 33.2 MB
    if (ws_size >= need) {
        float* ws = (float*)d_ws;
        const int npix = H_OUT * W_OUT;
        coolchic_up_pack_kernel<<<(npix + 255) / 256, 256, 0, stream>>>(
            g[0], g[1], g[2], g[3], g[4], g[5], g[6], g[7], ws);
        coolchic_sample_pack_kernel<<<(n + 255) / 256, 256, 0, stream>>>(x, ws, out, n);
    } else {
        coolchic_fused_kernel<<<(n + 255) / 256, 256, 0, stream>>>(
            x, g[0], g[1], g[2], g[3], g[4], g[5], g[6], g[7], out, n);
    }
}